// MakeMoreRNN_29781303230567
// MI455X (gfx1250) — compile-verified
//
#include <hip/hip_runtime.h>
#include <hip/hip_bf16.h>

#define VOCAB 27
#define EMB   32
#define HID   64
#define KC    (EMB + HID)   // 96
#define TLEN  256
#define BATCH 4096

#define XH_STRIDE 100       // 96 + 4 pad -> conflict-free ds_load_b64 columns
#define WC_STRIDE 100       // 96 + 4 pad
#define WH_STRIDE 68        // 64 + 4 pad

typedef __attribute__((ext_vector_type(2))) float v2f;
typedef __attribute__((ext_vector_type(4))) float v4f;
typedef __attribute__((ext_vector_type(8))) float v8f;

__launch_bounds__(128)
__global__ void rnn_fused_wmma_kernel(const int*   __restrict__ X,
                                      const float* __restrict__ emb_table,
                                      const float* __restrict__ start,
                                      const float* __restrict__ W_cell,
                                      const float* __restrict__ b_cell,
                                      const float* __restrict__ W_head,
                                      const float* __restrict__ b_head,
                                      float*       __restrict__ out)
{
    __shared__ float sWc[64 * WC_STRIDE];        // 25600 B, B-matrix source: Wc[N][K]
    __shared__ float sWh[32 * WH_STRIDE];        //  8704 B, head weights, rows 27..31 zero
    __shared__ float sXh[4][16 * XH_STRIDE];     // 25600 B, per-wave [x_t | h] panel

    const int tid  = threadIdx.x;
    const int wave = tid >> 5;
    const int lane = tid & 31;

    // ---- one-time: stage weights into LDS (shared by all 4 waves) ----
    for (int i = tid; i < 64 * KC; i += 128) {
        int r = i / KC, c = i - r * KC;
        sWc[r * WC_STRIDE + c] = W_cell[i];
    }
    for (int i = tid; i < 32 * WH_STRIDE; i += 128) sWh[i] = 0.0f;
    __syncthreads();
    for (int i = tid; i < VOCAB * HID; i += 128) {
        int r = i / HID, c = i - r * HID;
        sWh[r * WH_STRIDE + c] = W_head[i];
    }

    // ---- per-wave: init h panel with broadcast `start` ----
    float* xh = sXh[wave];
    for (int r = 0; r < 16; ++r)
        for (int c = lane; c < HID; c += 32)
            xh[r * XH_STRIDE + EMB + c] = start[c];
    __syncthreads();

    const int hi = lane >> 4;        // half-wave select (K pair / M+8)
    const int ln = lane & 15;        // N column / M row within tile
    const int rowbase = (blockIdx.x * 4 + wave) * 16;

    // bias fragments (C layout: every VGPR holds b[N=ln] of its N-tile)
    float cb0 = b_cell[ 0 + ln], cb1 = b_cell[16 + ln];
    float cb2 = b_cell[32 + ln], cb3 = b_cell[48 + ln];
    float hb0 = b_head[ln];
    float hb1 = (16 + ln < VOCAB) ? b_head[16 + ln] : 0.0f;

    // embedding gather: 2 lanes per batch row, 16 floats (half an emb row) each
    const int grow  = lane >> 1;
    const int ghalf = (lane & 1) * 16;

    for (int t = 0; t < TLEN; ++t) {
        // -------- gather x_t into xh[:, 0:32] (emb_table is L0/L2 resident) --------
        const int idx = X[(rowbase + grow) * TLEN + t];
        const float* erow = emb_table + idx * EMB + ghalf;
        float*       xrow = xh + grow * XH_STRIDE + ghalf;
        #pragma unroll
        for (int q = 0; q < 4; ++q) {
            v4f v = *(const v4f*)(erow + q * 4);
            *(v4f*)(xrow + q * 4) = v;
        }

        // -------- cell: h_new[16x64] = xh[16x96] @ Wc^T + b_cell --------
        v8f acc0 = {cb0, cb0, cb0, cb0, cb0, cb0, cb0, cb0};
        v8f acc1 = {cb1, cb1, cb1, cb1, cb1, cb1, cb1, cb1};
        v8f acc2 = {cb2, cb2, cb2, cb2, cb2, cb2, cb2, cb2};
        v8f acc3 = {cb3, cb3, cb3, cb3, cb3, cb3, cb3, cb3};
        #pragma unroll
        for (int kb = 0; kb < KC / 4; ++kb) {
            // A fragment 16x4: lanes 0-15 -> K=4kb..4kb+1, lanes 16-31 -> K=4kb+2..
            v2f a = *(const v2f*)(xh + ln * XH_STRIDE + kb * 4 + hi * 2);
            v2f b0 = *(const v2f*)(sWc + ( 0 + ln) * WC_STRIDE + kb * 4 + hi * 2);
            v2f b1 = *(const v2f*)(sWc + (16 + ln) * WC_STRIDE + kb * 4 + hi * 2);
            v2f b2 = *(const v2f*)(sWc + (32 + ln) * WC_STRIDE + kb * 4 + hi * 2);
            v2f b3 = *(const v2f*)(sWc + (48 + ln) * WC_STRIDE + kb * 4 + hi * 2);
            acc0 = __builtin_amdgcn_wmma_f32_16x16x4_f32(false, a, false, b0, (short)0, acc0, false, false);
            acc1 = __builtin_amdgcn_wmma_f32_16x16x4_f32(false, a, false, b1, (short)0, acc1, false, false);
            acc2 = __builtin_amdgcn_wmma_f32_16x16x4_f32(false, a, false, b2, (short)0, acc2, false, false);
            acc3 = __builtin_amdgcn_wmma_f32_16x16x4_f32(false, a, false, b3, (short)0, acc3, false, false);
        }

        // -------- write h_new back into xh[:, 32:96] (D layout: VGPR i -> M=i+8*hi, N=ln) --------
        #pragma unroll
        for (int i = 0; i < 8; ++i) {
            float* hrow = xh + (i + 8 * hi) * XH_STRIDE + EMB;
            hrow[ 0 + ln] = acc0[i];
            hrow[16 + ln] = acc1[i];
            hrow[32 + ln] = acc2[i];
            hrow[48 + ln] = acc3[i];
        }

        // -------- head: logits[16x27] = h_new[16x64] @ Wh^T + b_head --------
        v8f h0 = {hb0, hb0, hb0, hb0, hb0, hb0, hb0, hb0};
        v8f h1 = {hb1, hb1, hb1, hb1, hb1, hb1, hb1, hb1};
        #pragma unroll
        for (int kb = 0; kb < HID / 4; ++kb) {
            v2f a  = *(const v2f*)(xh + ln * XH_STRIDE + EMB + kb * 4 + hi * 2);
            v2f b0 = *(const v2f*)(sWh + ( 0 + ln) * WH_STRIDE + kb * 4 + hi * 2);
            v2f b1 = *(const v2f*)(sWh + (16 + ln) * WH_STRIDE + kb * 4 + hi * 2);
            h0 = __builtin_amdgcn_wmma_f32_16x16x4_f32(false, a, false, b0, (short)0, h0, false, false);
            h1 = __builtin_amdgcn_wmma_f32_16x16x4_f32(false, a, false, b1, (short)0, h1, false, false);
        }

        // -------- store logits (lanes 0-15 of each D VGPR are 64B-contiguous) --------
        #pragma unroll
        for (int i = 0; i < 8; ++i) {
            const int row = rowbase + i + 8 * hi;
            float* orow = out + ((size_t)row * TLEN + t) * VOCAB;
            orow[ln] = h0[i];
            if (16 + ln < VOCAB) orow[16 + ln] = h1[i];
        }
    }
}

extern "C" void kernel_launch(void* const* d_in, const int* in_sizes, int n_in,
                              void* d_out, int out_size, void* d_ws, size_t ws_size,
                              hipStream_t stream) {
    const int*   X         = (const int*)d_in[0];
    const float* emb_table = (const float*)d_in[1];
    const float* start     = (const float*)d_in[2];
    const float* W_cell    = (const float*)d_in[3];
    const float* b_cell    = (const float*)d_in[4];
    const float* W_head    = (const float*)d_in[5];
    const float* b_head    = (const float*)d_in[6];
    float* out = (float*)d_out;

    // 64 blocks x 128 threads = 256 waves; each wave owns 16 batch rows.
    dim3 grid(BATCH / (16 * 4));
    dim3 block(128);
    rnn_fused_wmma_kernel<<<grid, block, 0, stream>>>(
        X, emb_table, start, W_cell, b_cell, W_head, b_head, out);
}